// VectorQuantizerEMA_30631706755895
// MI455X (gfx1250) — compile-verified
//
#include <hip/hip_runtime.h>
#include <hip/hip_bf16.h>
#include <stdint.h>

// Problem constants
#define BB 32
#define SS 1024
#define DD 256
#define NN 1024
#define ROWS (BB * SS)          // 32768
#define QELEMS (ROWS * DD)      // 8388608
#define JT_N 32                 // emb rows per staged tile
#define MT 64                   // input rows per workgroup

typedef float v2f __attribute__((ext_vector_type(2)));
typedef float v8f __attribute__((ext_vector_type(8)));
typedef unsigned int u32x4 __attribute__((ext_vector_type(4)));
typedef int i32x8 __attribute__((ext_vector_type(8)));
typedef int i32x4 __attribute__((ext_vector_type(4)));

// ---- Tensor Data Mover helpers (D# per CDNA5 ISA ch.8) -------------------
// group0: [1:0]=count=1, [63:32]=lds_addr, [120:64]=global_addr, [127:126]=type=2
// group1: [17:16]=data_size(2 -> 4B), tensor_dim0/1, tile_dim0/1, dim0 stride
__device__ __forceinline__ unsigned lds_offset(const void* p) {
  return (unsigned)(unsigned long long)(uintptr_t)p;   // aperture: addr[31:0] = LDS offset
}

__device__ __forceinline__ void tdm_load_2d(unsigned lds_off, const void* gptr,
                                            unsigned tile_d0, unsigned tile_d1,
                                            unsigned tensor_d0, unsigned tensor_d1,
                                            unsigned stride0) {
  unsigned long long ga = (unsigned long long)(uintptr_t)gptr;
  u32x4 g0 = { 1u, lds_off, (unsigned)ga,
               (unsigned)((ga >> 32) & 0x01FFFFFFu) | 0x80000000u };
  i32x8 g1 = { (int)0x20000u,                                   // data_size = 4B
               (int)((tensor_d0 & 0xFFFFu) << 16),              // tensor_dim0 lo
               (int)((tensor_d0 >> 16) | ((tensor_d1 & 0xFFFFu) << 16)),
               (int)((tensor_d1 >> 16) | (tile_d0 << 16)),      // tile_dim0
               (int)tile_d1,                                    // tile_dim1 (tile_dim2=0)
               (int)stride0,                                    // tensor_dim0_stride lo
               0, 0 };
  i32x4 z4 = { 0, 0, 0, 0 };
  i32x8 z8 = { 0, 0, 0, 0, 0, 0, 0, 0 };
  __builtin_amdgcn_tensor_load_to_lds(g0, g1, z4, z4, z8, 0);
}

__device__ __forceinline__ void tdm_load_1d(unsigned lds_off, const void* gptr,
                                            unsigned tile_elems, unsigned tensor_elems) {
  unsigned long long ga = (unsigned long long)(uintptr_t)gptr;
  u32x4 g0 = { 1u, lds_off, (unsigned)ga,
               (unsigned)((ga >> 32) & 0x01FFFFFFu) | 0x80000000u };
  i32x8 g1 = { (int)0x20000u,
               (int)((tensor_elems & 0xFFFFu) << 16),
               (int)(tensor_elems >> 16),
               (int)(tile_elems << 16),                         // tile_dim0 (1D)
               0,
               (int)tensor_elems,
               0, 0 };
  i32x4 z4 = { 0, 0, 0, 0 };
  i32x8 z8 = { 0, 0, 0, 0, 0, 0, 0, 0 };
  __builtin_amdgcn_tensor_load_to_lds(g0, g1, z4, z4, z8, 0);
}

// ---------------- kernel 0: embedding squared norms ----------------
__global__ __launch_bounds__(64) void k_esq(const float* __restrict__ e,
                                            float* __restrict__ esq) {
  __shared__ float p[64];
  int j = blockIdx.x;
  float s = 0.f;
  for (int k = threadIdx.x; k < DD; k += 64) {
    float v = e[j * DD + k];
    s += v * v;
  }
  p[threadIdx.x] = s;
  __syncthreads();
  for (int o = 32; o > 0; o >>= 1) {
    if (threadIdx.x < o) p[threadIdx.x] += p[threadIdx.x + o];
    __syncthreads();
  }
  if (threadIdx.x == 0) esq[j] = p[0];
}

// ---------------- kernel 1: fused GEMM distance + min/argmin ----------------
// 512 blocks x 128 threads (4 waves). Block owns MT=64 rows.
// TDM stages x-tile once and double-buffers 32-row emb tiles (TENSORcnt pipeline).
// Each wave computes a 16x32 tile per stage: 2 WMMAs share one A-fragment.
__global__ __launch_bounds__(128) void k_dist(const float* __restrict__ x,
                                              const float* __restrict__ e,
                                              const float* __restrict__ esq,
                                              float* __restrict__ dmin,
                                              int* __restrict__ amin) {
  __shared__ float xt[MT * DD];            // 64KB x-tile
  __shared__ float et[2 * JT_N * DD];      // 2 x 32KB emb tile double buffer
  __shared__ float xsq_p[128];
  __shared__ float xsq_s[MT];

  const int tid = threadIdx.x;
  const int wave = tid >> 5;
  const int lane = tid & 31;
  const int l15 = lane & 15;
  const int halfsel = lane >> 4;           // 0: lanes 0-15, 1: lanes 16-31
  const long rowbase = (long)blockIdx.x * MT;

  // Issue TDM: x-tile (2D: 256 x 64 rows), then first emb tile (1D contiguous)
  if (tid < 32) {
    tdm_load_2d(lds_offset(xt), x + rowbase * DD,
                /*tile*/ DD, MT, /*tensor*/ DD, ROWS, /*stride0*/ DD);
    tdm_load_1d(lds_offset(et), e, JT_N * DD, NN * DD);
    __builtin_amdgcn_s_wait_tensorcnt(1);  // x-tile complete (in-order)
  }
  __syncthreads();

  // Row squared norms from x-tile: 2 threads per row, 128 cols each
  {
    int r = tid & 63;
    int half = tid >> 6;
    const float* xr = xt + r * DD + half * 128;
    float s = 0.f;
    for (int k = 0; k < 128; ++k) s += xr[k] * xr[k];
    xsq_p[tid] = s;
    __syncthreads();
    if (tid < MT) xsq_s[tid] = xsq_p[tid] + xsq_p[tid + 64];
    __syncthreads();
  }

  const int mrow0 = wave * 16;             // this wave's 16-row strip
  // A-fragment (16x4 f32): lanes 0-15 hold M=l,K=k0..k0+1; lanes 16-31 K=k0+2..k0+3
  const float* aptr = xt + (mrow0 + l15) * DD + halfsel * 2;

  float xsqr[8];
#pragma unroll
  for (int r = 0; r < 8; ++r) xsqr[r] = xsq_s[mrow0 + r + 8 * halfsel];

  unsigned long long bestkey[8];
#pragma unroll
  for (int r = 0; r < 8; ++r) bestkey[r] = ~0ull;

  for (int jt = 0; jt < NN / JT_N; ++jt) {
    if (tid < 32) __builtin_amdgcn_s_wait_tensorcnt(0);   // tile jt landed
    __syncthreads();   // publish tile jt; all waves done reading buffer (jt+1)&1
    if (tid < 32 && jt + 1 < NN / JT_N) {
      tdm_load_1d(lds_offset(et + ((jt + 1) & 1) * (JT_N * DD)),
                  e + (long)(jt + 1) * JT_N * DD, JT_N * DD, NN * DD);
    }

    const float* bbase = et + (jt & 1) * (JT_N * DD);
    const float* b0 = bbase + l15 * DD + halfsel * 2;          // cols jt*32 + 0..15
    const float* b1 = bbase + (16 + l15) * DD + halfsel * 2;   // cols jt*32 + 16..31

    v8f acc0 = {};
    v8f acc1 = {};
#pragma unroll 8
    for (int k0 = 0; k0 < DD; k0 += 4) {
      v2f a  = *(const v2f*)(aptr + k0);
      v2f vb0 = *(const v2f*)(b0 + k0);
      v2f vb1 = *(const v2f*)(b1 + k0);
      acc0 = __builtin_amdgcn_wmma_f32_16x16x4_f32(false, a, false, vb0,
                                                   (short)0, acc0, false, false);
      acc1 = __builtin_amdgcn_wmma_f32_16x16x4_f32(false, a, false, vb1,
                                                   (short)0, acc1, false, false);
    }

    const int j0 = jt * JT_N + l15;
    const int j1 = j0 + 16;
    const float e0 = esq[j0];
    const float e1 = esq[j1];
#pragma unroll
    for (int r = 0; r < 8; ++r) {
      float dv0 = xsqr[r] + e0 - 2.0f * acc0[r];   // >= 0: bit order == float order
      float dv1 = xsqr[r] + e1 - 2.0f * acc1[r];
      unsigned long long k0v =
          ((unsigned long long)__float_as_uint(dv0) << 32) | (unsigned)j0;
      unsigned long long k1v =
          ((unsigned long long)__float_as_uint(dv1) << 32) | (unsigned)j1;
      if (k1v < k0v) k0v = k1v;
      if (k0v < bestkey[r]) bestkey[r] = k0v;
    }
  }

  // min-reduce keys across the 16 lanes of each half (rows stay separated)
#pragma unroll
  for (int off = 1; off < 16; off <<= 1) {
#pragma unroll
    for (int r = 0; r < 8; ++r) {
      unsigned long long o = __shfl_xor(bestkey[r], off, 32);
      if (o < bestkey[r]) bestkey[r] = o;
    }
  }

  // lanes 0-7 write rows mrow0+0..7, lanes 16-23 write rows mrow0+8..15
  if (l15 < 8) {
    int r = l15;
    int m = mrow0 + r + 8 * halfsel;
    unsigned long long k = bestkey[r];
    dmin[rowbase + m] = __uint_as_float((unsigned)(k >> 32));
    amin[rowbase + m] = (int)(unsigned)k;
  }
}

// ---------------- kernel 2: per-batch stable argsort + faithful reorder ----------------
__global__ __launch_bounds__(512) void k_sort(const float* __restrict__ dmin,
                                              const int* __restrict__ amin,
                                              int* __restrict__ fidx,
                                              float* __restrict__ idx_out,
                                              unsigned int* __restrict__ counts) {
  __shared__ unsigned long long keys[SS];
  const int b = blockIdx.x;
  const int tid = threadIdx.x;

  for (int i = tid; i < SS; i += 512) {
    float dv = dmin[b * SS + i];
    keys[i] = ((unsigned long long)__float_as_uint(dv) << 32) | (unsigned)i;
  }
  __syncthreads();

  for (int ksz = 2; ksz <= SS; ksz <<= 1) {
    for (int jsz = ksz >> 1; jsz > 0; jsz >>= 1) {
      for (int i = tid; i < SS; i += 512) {
        int l = i ^ jsz;
        if (l > i) {
          unsigned long long a = keys[i], c = keys[l];
          bool asc = ((i & ksz) == 0);
          if ((a > c) == asc) { keys[i] = c; keys[l] = a; }
        }
      }
      __syncthreads();
    }
  }

  // Faithful bug: enc_idx_new[b*S+rank] = enc_idx_old[order] -> indexes batch 0!
  for (int rank = tid; rank < SS; rank += 512) {
    int src = (int)(unsigned)keys[rank];   // order value in [0, S) -> batch-0 slot
    int id = amin[src];
    fidx[b * SS + rank] = id;
    idx_out[b * SS + rank] = (float)id;
    atomicAdd(&counts[id], 1u);
  }
}

// ---------------- kernel 3: gather + straight-through output + SSE partials ----------------
__global__ __launch_bounds__(256) void k_gather(const float* __restrict__ x,
                                                const float* __restrict__ e,
                                                const int* __restrict__ fidx,
                                                float* __restrict__ qout,
                                                float* __restrict__ ssep) {
  __shared__ float r8[8];
  const int row = blockIdx.x;
  const int c = threadIdx.x;
  const int wv = c >> 5;
  const int lane = c & 31;

  int id = fidx[row];
  float q = e[id * DD + c];
  float xin = x[(long)row * DD + c];
  float diff = q - xin;
  qout[(long)row * DD + c] = xin + diff;   // inputs + stop_grad(quantized - inputs)

  float v = diff * diff;
#pragma unroll
  for (int o = 16; o > 0; o >>= 1) v += __shfl_down(v, o, 32);
  if (lane == 0) r8[wv] = v;
  __syncthreads();
  if (c == 0) {
    float t = 0.f;
    for (int w = 0; w < 8; ++w) t += r8[w];
    ssep[row] = t;
  }
}

// ---------------- kernel 4: deterministic finalize (loss, perplexity) ----------------
__global__ __launch_bounds__(1024) void k_final(const unsigned int* __restrict__ counts,
                                                const float* __restrict__ ssep,
                                                float* __restrict__ loss_out,
                                                float* __restrict__ ppl_out) {
  __shared__ float ra[1024];
  __shared__ float rb[1024];
  const int t = threadIdx.x;

  float s = 0.f;
  for (int k = 0; k < ROWS / 1024; ++k) s += ssep[t + (k << 10)];
  ra[t] = s;

  float p = (float)counts[t] * (1.0f / (float)ROWS);
  rb[t] = p * logf(p + 1e-10f);
  __syncthreads();
  for (int o = 512; o > 0; o >>= 1) {
    if (t < o) { ra[t] += ra[t + o]; rb[t] += rb[t + o]; }
    __syncthreads();
  }
  if (t == 0) {
    loss_out[0] = ra[0] * (1.25f / (float)QELEMS);  // q_latent + 0.25*e_latent
    ppl_out[0] = expf(-rb[0]);
  }
}

// ---------------- launch ----------------
extern "C" void kernel_launch(void* const* d_in, const int* in_sizes, int n_in,
                              void* d_out, int out_size, void* d_ws, size_t ws_size,
                              hipStream_t stream) {
  (void)in_sizes; (void)n_in; (void)out_size; (void)ws_size;
  const float* x = (const float*)d_in[0];   // [B,S,D]
  const float* e = (const float*)d_in[1];   // [N,D]
  float* out = (float*)d_out;

  char* ws = (char*)d_ws;
  float*        esq    = (float*)(ws + 0);                   // 1024 f32
  float*        dmin   = (float*)(ws + 4096);                // 32768 f32
  int*          amin   = (int*)(ws + 135168);                // 32768 i32
  int*          fidx   = (int*)(ws + 266240);                // 32768 i32
  unsigned int* counts = (unsigned int*)(ws + 397312);       // 1024 u32
  float*        ssep   = (float*)(ws + 401408);              // 32768 f32

  float* q_out   = out + 1;                 // [B,S,D]
  float* ppl_out = out + 1 + QELEMS;        // scalar
  float* idx_out = out + 2 + QELEMS;        // [B,S] as float

  (void)hipMemsetAsync(counts, 0, NN * sizeof(unsigned int), stream);

  k_esq   <<<NN, 64, 0, stream>>>(e, esq);
  k_dist  <<<ROWS / MT, 128, 0, stream>>>(x, e, esq, dmin, amin);
  k_sort  <<<BB, 512, 0, stream>>>(dmin, amin, fidx, idx_out, counts);
  k_gather<<<ROWS, 256, 0, stream>>>(x, e, fidx, q_out, ssep);
  k_final <<<1, 1024, 0, stream>>>(counts, ssep, out, ppl_out);
}